// LLRPrior_42460046688256
// MI455X (gfx1250) — compile-verified
//
#include <hip/hip_runtime.h>

// LLR prior: per 8x8 patch, 32x64 Casorati matrix SVD soft-threshold.
//   G = M M^T (32x32)               -> v_wmma_f32_16x16x4_f32
//   G = U S^2 U^T via parallel Jacobi (VALU, closed-form round-robin schedule)
//   Y = diag(soft(s)/s) * (U^T M)   -> v_wmma_f32_16x16x4_f32
//   out = U Y                        -> v_wmma_f32_16x16x4_f32

#define BATCH 8
#define TDIM 32          // contrast axis T (matrix rows)
#define HDIM 384
#define WDIM 384
#define PSZ  8
#define NH   (HDIM / PSZ)   // 48
#define NW   (WDIM / PSZ)   // 48
#define KC   (PSZ * PSZ)    // 64 (matrix cols)
#define THS  0.1f
#define NSWEEP 8

typedef __attribute__((ext_vector_type(2))) float v2f;
typedef __attribute__((ext_vector_type(8))) float v8f;

__device__ __forceinline__ v8f wmma4(v2f a, v2f b, v8f c) {
  // D = A(16x4,f32) * B(4x16,f32) + C(16x16,f32)
  return __builtin_amdgcn_wmma_f32_16x16x4_f32(
      /*neg_a=*/false, a, /*neg_b=*/false, b,
      /*c_mod=*/(short)0, c, /*reuse_a=*/false, /*reuse_b=*/false);
}

// Round-robin (circle method) pairing: player 31 fixed, others rotate.
// Round rr in [0,30], pair k in [0,15]; every round is a perfect matching,
// all 496 pairs are covered per 31-round sweep.
__device__ __forceinline__ void jacobi_pair(int rr, int k, int& p, int& q) {
  if (k == 0) {
    p = 31;
    q = rr;
  } else {
    p = (rr + k) % 31;
    q = (rr + 31 - k) % 31;
  }
}

__global__ __launch_bounds__(128) void llr_svt_kernel(const float* __restrict__ x,
                                                      float* __restrict__ out) {
  __shared__ float Mlds[TDIM][KC + 1];      // 32 x 65 (odd stride -> conflict-free cols)
  __shared__ float Ylds[TDIM][KC + 1];
  __shared__ float G[TDIM][TDIM + 1];       // 32 x 33
  __shared__ float U[TDIM][TDIM + 1];
  __shared__ float wsv[TDIM];

  const int tid  = threadIdx.x;
  const int lane = tid & 31;
  const int wave = tid >> 5;
  const int lm   = lane & 15;   // M/N index within a 16x16 tile
  const int lh   = lane >> 4;   // half-wave select

  const int pid = blockIdx.x;
  const int b   = pid / (NH * NW);
  const int pr  = pid % (NH * NW);
  const int ph  = pr / NW;
  const int pw  = pr % NW;

  const size_t base = (size_t)b * TDIM * HDIM * WDIM
                    + (size_t)(ph * PSZ) * WDIM + (size_t)(pw * PSZ);

  // ---- gather patch -> M (32x64), coalesced b128 loads ----
  for (int j = 0; j < 4; ++j) {
    const int e   = tid + j * 128;    // float4 index 0..511
    const int lin = e * 4;
    const int t   = lin >> 6;
    const int col = lin & 63;         // multiple of 4
    const int p   = col >> 3;
    const int q   = col & 7;          // 0 or 4: stays inside one 8-wide patch row
    const float4 v = *reinterpret_cast<const float4*>(
        x + base + ((size_t)t * HDIM + p) * WDIM + q);
    Mlds[t][col + 0] = v.x;
    Mlds[t][col + 1] = v.y;
    Mlds[t][col + 2] = v.z;
    Mlds[t][col + 3] = v.w;
  }
  // init U = I
  for (int e = tid; e < TDIM * TDIM; e += 128) {
    const int i = e >> 5, jj = e & 31;
    U[i][jj] = (i == jj) ? 1.0f : 0.0f;
  }
  __syncthreads();

  // ---- G = M M^T : four 16x16 tiles, one per wave ----
  {
    const int tm = (wave >> 1) * 16;
    const int tn = (wave & 1) * 16;
    v8f acc = {0.f, 0.f, 0.f, 0.f, 0.f, 0.f, 0.f, 0.f};
    for (int kk = 0; kk < KC; kk += 4) {
      const int k0 = kk + 2 * lh;
      v2f a, bb;
      a.x  = Mlds[tm + lm][k0];
      a.y  = Mlds[tm + lm][k0 + 1];
      bb.x = Mlds[tn + lm][k0];        // B[k][n] = M[n][k]
      bb.y = Mlds[tn + lm][k0 + 1];
      acc = wmma4(a, bb, acc);
    }
    for (int v = 0; v < 8; ++v)
      G[tm + v + 8 * lh][tn + lm] = acc[v];
  }
  __syncthreads();

  // ---- parallel cyclic Jacobi ----
  // Wave w owns pairs {w, w+4, w+8, w+12}: wave-uniform p,q,c,s; i/j = lane
  // gives conflict-free LDS column/row access. Coefficient reads live in the
  // pair's own columns (which only that pair writes), so they are computed
  // inside phase A and carried in registers to phase B: 2 barriers/round.
  for (int r = 0; r < 31 * NSWEEP; ++r) {
    const int rr = r % 31;
    float cbuf[4], sbuf[4];
    // phase A: G <- G*J (columns), U <- U*J
#pragma unroll
    for (int j = 0; j < 4; ++j) {
      const int k = wave + j * 4;
      int p, q;
      jacobi_pair(rr, k, p, q);
      const float app = G[p][p], aqq = G[q][q], apq = G[p][q];
      float c = 1.0f, s = 0.0f;
      if (fabsf(apq) > 1e-12f) {
        const float tau = (aqq - app) / (2.0f * apq);
        const float t = ((tau >= 0.0f) ? 1.0f : -1.0f) /
                        (fabsf(tau) + sqrtf(1.0f + tau * tau));
        c = 1.0f / sqrtf(1.0f + t * t);
        s = t * c;
      }
      cbuf[j] = c; sbuf[j] = s;
      const int i = lane;
      const float gp = G[i][p], gq = G[i][q];
      G[i][p] = c * gp - s * gq;
      G[i][q] = s * gp + c * gq;
      const float up = U[i][p], uq = U[i][q];
      U[i][p] = c * up - s * uq;
      U[i][q] = s * up + c * uq;
    }
    __syncthreads();
    // phase B: G <- J^T*G (rows)
#pragma unroll
    for (int j = 0; j < 4; ++j) {
      const int k = wave + j * 4;
      int p, q;
      jacobi_pair(rr, k, p, q);
      const float c = cbuf[j], s = sbuf[j];
      const int jc = lane;
      const float gp = G[p][jc], gq = G[q][jc];
      G[p][jc] = c * gp - s * gq;
      G[q][jc] = s * gp + c * gq;
    }
    __syncthreads();
  }

  // ---- singular values & soft-threshold weights: w_i = soft(s_i)/s_i ----
  if (tid < TDIM) {
    const float g  = G[tid][tid];
    const float sv = sqrtf(fmaxf(g, 0.0f));
    const float st = (sv > THS) ? (sv - THS) : 0.0f;  // s >= 0 always
    wsv[tid] = (sv > 1e-12f) ? (st / sv) : 0.0f;
  }
  __syncthreads();

  // ---- Y = diag(w) * (U^T M) : eight 16x16 tiles, two per wave ----
  for (int ti = wave; ti < 8; ti += 4) {
    const int tm = (ti >> 2) * 16;
    const int tn = (ti & 3) * 16;
    v8f acc = {0.f, 0.f, 0.f, 0.f, 0.f, 0.f, 0.f, 0.f};
    for (int kk = 0; kk < TDIM; kk += 4) {
      const int k0 = kk + 2 * lh;
      v2f a, bb;
      a.x  = U[k0][tm + lm];          // A[m][k] = U[k][m]  (U^T)
      a.y  = U[k0 + 1][tm + lm];
      bb.x = Mlds[k0][tn + lm];
      bb.y = Mlds[k0 + 1][tn + lm];
      acc = wmma4(a, bb, acc);
    }
    for (int v = 0; v < 8; ++v) {
      const int rrow = tm + v + 8 * lh;
      Ylds[rrow][tn + lm] = acc[v] * wsv[rrow];
    }
  }
  __syncthreads();

  // ---- out = U * Y : reuse Mlds as the output staging buffer ----
  for (int ti = wave; ti < 8; ti += 4) {
    const int tm = (ti >> 2) * 16;
    const int tn = (ti & 3) * 16;
    v8f acc = {0.f, 0.f, 0.f, 0.f, 0.f, 0.f, 0.f, 0.f};
    for (int kk = 0; kk < TDIM; kk += 4) {
      const int k0 = kk + 2 * lh;
      v2f a, bb;
      a.x  = U[tm + lm][k0];
      a.y  = U[tm + lm][k0 + 1];
      bb.x = Ylds[k0][tn + lm];
      bb.y = Ylds[k0 + 1][tn + lm];
      acc = wmma4(a, bb, acc);
    }
    for (int v = 0; v < 8; ++v)
      Mlds[tm + v + 8 * lh][tn + lm] = acc[v];
  }
  __syncthreads();

  // ---- scatter patch back, coalesced b128 stores ----
  for (int j = 0; j < 4; ++j) {
    const int e   = tid + j * 128;
    const int lin = e * 4;
    const int t   = lin >> 6;
    const int col = lin & 63;
    const int p   = col >> 3;
    const int q   = col & 7;
    float4 v;
    v.x = Mlds[t][col + 0];
    v.y = Mlds[t][col + 1];
    v.z = Mlds[t][col + 2];
    v.w = Mlds[t][col + 3];
    *reinterpret_cast<float4*>(out + base + ((size_t)t * HDIM + p) * WDIM + q) = v;
  }
}

extern "C" void kernel_launch(void* const* d_in, const int* in_sizes, int n_in,
                              void* d_out, int out_size, void* d_ws, size_t ws_size,
                              hipStream_t stream) {
  (void)in_sizes; (void)n_in; (void)out_size; (void)d_ws; (void)ws_size;
  const float* x = (const float*)d_in[0];
  float* o = (float*)d_out;
  const int nblocks = BATCH * NH * NW;   // 18432 matrices, one block each
  llr_svt_kernel<<<dim3(nblocks), dim3(128), 0, stream>>>(x, o);
}